// VectorQuantizer_30666066493667
// MI455X (gfx1250) — compile-verified
//
#include <hip/hip_runtime.h>
#include <hip/hip_bf16.h>

typedef float v2f __attribute__((ext_vector_type(2)));
typedef float v4f __attribute__((ext_vector_type(4)));
typedef float v8f __attribute__((ext_vector_type(8)));

// ---------------------------------------------------------------------------
// Kernel 1: codebook row norms ||c_k||^2  (K rows of 64 floats; one warp/row)
// ---------------------------------------------------------------------------
__global__ __launch_bounds__(256) void vq_csq_kernel(const float* __restrict__ cb,
                                                     float* __restrict__ csq, int K) {
    const int w = threadIdx.x >> 5;
    const int lane = threadIdx.x & 31;
    const int row = blockIdx.x * 8 + w;
    if (row >= K) return;
    const v2f* r = (const v2f*)(cb + (size_t)row * 64);
    v2f c = r[lane];
    float p = c.x * c.x + c.y * c.y;
#pragma unroll
    for (int off = 16; off; off >>= 1) p += __shfl_xor(p, off, 32);
    if (lane == 0) csq[row] = p;
}

// ---------------------------------------------------------------------------
// Kernel 2: fused GEMM (f32 WMMA) + argmin + gather + loss partials.
// Block = 256 threads = 8 waves; 128 rows of z per block.
// - A tile (16x64) per wave in registers (ISA 32-bit A 16x4 layout).
// - Codebook chunk stored in LDS as (d,d+1) float2 PAIRS per column, so each
//   WMMA B operand is one aligned ds_load_b64 (no register marshalling).
//   Pair-row stride 144 float2 => the two half-wave rows are 32 banks apart.
// - Double-buffered LDS with register prefetch overlaps staging with WMMA.
// - 4 independent WMMA accumulator chains (2 N-tiles x 2 K-halves) for ILP.
// ---------------------------------------------------------------------------
__global__ __launch_bounds__(256) void vq_main_kernel(const float* __restrict__ z,
                                                      const float* __restrict__ cb,
                                                      const float* __restrict__ csq,
                                                      float* __restrict__ out_zq,
                                                      float* __restrict__ out_idx,
                                                      float* __restrict__ partials,
                                                      int K) {
    constexpr int D    = 64;    // embedding dim
    constexpr int MB   = 128;   // rows per block
    constexpr int NC   = 128;   // codes per LDS chunk
    constexpr int DP   = D / 2; // d-pair rows
    constexpr int STRP = 144;   // float2 stride per pair-row
    constexpr int KMAX = 4096;  // csq table staged in LDS

    __shared__ v2f   sCbP[2 * DP * STRP];  // double-buffered paired chunk
    __shared__ float sCsq[KMAX];
    __shared__ int   sIdx[MB];
    __shared__ float sLoss[8];

    const int tid    = threadIdx.x;
    const int lane   = tid & 31;
    const int w      = tid >> 5;
    const int ln15   = lane & 15;
    const bool hiHalf = (lane >= 16);
    const int kh     = hiHalf ? 1 : 0;     // pair-row sub-offset (K sub 0/2)
    const int rbase  = hiHalf ? 8 : 0;     // C/D row base (32-bit C layout)
    const int mBlock = blockIdx.x * MB;

    // Staging mapping: thread covers code kl = tid>>1, dims [d0, d0+32)
    const int klS = tid >> 1;
    const int d0S = (tid & 1) * 32;
    const int p0S = d0S >> 1;              // starting pair-row

    // ---- A tile in registers ----
    const int rowA = mBlock + w * 16 + ln15;
    const v2f* zrow = (const v2f*)(z + (size_t)rowA * D);
    v2f a[16];
#pragma unroll
    for (int kg = 0; kg < 16; ++kg)
        a[kg] = zrow[kg * 2 + kh];         // floats [kg*4+2*kh, kg*4+2*kh+1]

    // ---- ||z_row||^2 from A registers ----
    float p = 0.f;
#pragma unroll
    for (int kg = 0; kg < 16; ++kg) p += a[kg].x * a[kg].x + a[kg].y * a[kg].y;
    const float zr = p + __shfl_xor(p, 16, 32);
    float zsqv[8];
#pragma unroll
    for (int r = 0; r < 8; ++r) zsqv[r] = __shfl(zr, rbase + r, 32);

    float best[8];
    int   bidx[8];
#pragma unroll
    for (int r = 0; r < 8; ++r) { best[r] = __builtin_inff(); bidx[r] = 0; }

    // ---- one-time: csq table to LDS + stage chunk 0 into buffer 0 ----
    for (int i = tid; i < K; i += 256) sCsq[i] = csq[i];
    {
        const v4f* src = (const v4f*)(cb + (size_t)klS * D + d0S);
        v4f s[8];
#pragma unroll
        for (int j = 0; j < 8; ++j) s[j] = src[j];
#pragma unroll
        for (int j = 0; j < 8; ++j) {
            v2f lo = {s[j][0], s[j][1]};
            v2f hi = {s[j][2], s[j][3]};
            sCbP[(p0S + j * 2) * STRP + klS]     = lo;
            sCbP[(p0S + j * 2 + 1) * STRP + klS] = hi;
        }
    }
    __syncthreads();

    const int NCH = K / NC;
    for (int kc = 0; kc < NCH; ++kc) {
        const v2f* buf = sCbP + (kc & 1) * (DP * STRP);
        const int k0 = kc * NC;

        // prefetch next chunk into registers (overlaps the WMMA chains)
        v4f stg[8];
        const bool pf = (kc + 1 < NCH);
        if (pf) {
            const v4f* src =
                (const v4f*)(cb + (size_t)((kc + 1) * NC + klS) * D + d0S);
#pragma unroll
            for (int j = 0; j < 8; ++j) stg[j] = src[j];
        }

        // ---- compute: 4 independent WMMA chains (2 N-tiles x 2 K-halves) ----
        for (int nb = 0; nb < NC / 16; nb += 2) {
            const int n0 = nb * 16;
            const int n1 = n0 + 16;
            const v2f* col0 = buf + n0 + ln15;
            const v2f* col1 = buf + n1 + ln15;
            v8f acc0a = {0.f,0.f,0.f,0.f,0.f,0.f,0.f,0.f};
            v8f acc0b = {0.f,0.f,0.f,0.f,0.f,0.f,0.f,0.f};
            v8f acc1a = {0.f,0.f,0.f,0.f,0.f,0.f,0.f,0.f};
            v8f acc1b = {0.f,0.f,0.f,0.f,0.f,0.f,0.f,0.f};
#pragma unroll
            for (int kg = 0; kg < 8; ++kg) {
                const int prA = (kg * 2 + kh) * STRP;        // K-half A pair-row
                const int prB = ((kg + 8) * 2 + kh) * STRP;  // K-half B pair-row
                const v2f b0a = col0[prA];   // single ds_load_b64 each
                const v2f b1a = col1[prA];
                const v2f b0b = col0[prB];
                const v2f b1b = col1[prB];
                acc0a = __builtin_amdgcn_wmma_f32_16x16x4_f32(
                    false, a[kg],     false, b0a, (short)0, acc0a, false, false);
                acc1a = __builtin_amdgcn_wmma_f32_16x16x4_f32(
                    false, a[kg],     false, b1a, (short)0, acc1a, false, false);
                acc0b = __builtin_amdgcn_wmma_f32_16x16x4_f32(
                    false, a[kg + 8], false, b0b, (short)0, acc0b, false, false);
                acc1b = __builtin_amdgcn_wmma_f32_16x16x4_f32(
                    false, a[kg + 8], false, b1b, (short)0, acc1b, false, false);
            }
            const v8f sum0 = acc0a + acc0b;
            const v8f sum1 = acc1a + acc1b;
            {
                const int   code = k0 + n0 + ln15;
                const float cs   = sCsq[code];
#pragma unroll
                for (int r = 0; r < 8; ++r) {
                    const float dist = zsqv[r] - 2.0f * sum0[r] + cs;
                    if (dist < best[r]) { best[r] = dist; bidx[r] = code; }
                }
            }
            {
                const int   code = k0 + n1 + ln15;
                const float cs   = sCsq[code];
#pragma unroll
                for (int r = 0; r < 8; ++r) {
                    const float dist = zsqv[r] - 2.0f * sum1[r] + cs;
                    if (dist < best[r]) { best[r] = dist; bidx[r] = code; }
                }
            }
        }

        // drain prefetched registers into the other LDS buffer
        if (pf) {
            v2f* buf2 = sCbP + ((kc + 1) & 1) * (DP * STRP);
#pragma unroll
            for (int j = 0; j < 8; ++j) {
                v2f lo = {stg[j][0], stg[j][1]};
                v2f hi = {stg[j][2], stg[j][3]};
                buf2[(p0S + j * 2) * STRP + klS]     = lo;
                buf2[(p0S + j * 2 + 1) * STRP + klS] = hi;
            }
        }
        __syncthreads();
    }

    // ---- argmin across the 16 lanes of each half (tie -> smaller index) ----
#pragma unroll
    for (int r = 0; r < 8; ++r) {
        float bd = best[r];
        int   bi = bidx[r];
#pragma unroll
        for (int off = 1; off < 16; off <<= 1) {
            const float od = __shfl_xor(bd, off, 32);
            const int   oi = __shfl_xor(bi, off, 32);
            if (od < bd || (od == bd && oi < bi)) { bd = od; bi = oi; }
        }
        if (ln15 == 0) sIdx[w * 16 + rbase + r] = bi;
    }
    __syncthreads();

    // ---- gather z_q, write outputs, accumulate loss partial ----
    float lsum = 0.f;
#pragma unroll 4
    for (int i = 0; i < (MB * D) / 256; ++i) {
        const int e  = i * 256 + tid;
        const int rl = e >> 6;
        const int d  = e & 63;
        const int gi = sIdx[rl];
        const float v = cb[(size_t)gi * D + d];
        const size_t go = (size_t)(mBlock + rl) * D + d;
        out_zq[go] = v;                       // z_q_st forward value == z_q
        const float df = z[go] - v;
        lsum += df * df;
    }
    if (tid < MB) out_idx[mBlock + tid] = (float)sIdx[tid];

#pragma unroll
    for (int off = 16; off; off >>= 1) lsum += __shfl_xor(lsum, off, 32);
    if (lane == 0) sLoss[w] = lsum;
    __syncthreads();
    if (w == 0) {
        float v = (lane < 8) ? sLoss[lane] : 0.f;
#pragma unroll
        for (int off = 4; off; off >>= 1) v += __shfl_xor(v, off, 32);
        if (lane == 0) partials[blockIdx.x] = v;   // deterministic per-block sum
    }
}

// ---------------------------------------------------------------------------
// Kernel 3: deterministic final loss reduction (fixed-order strided sum)
// ---------------------------------------------------------------------------
__global__ void vq_loss_kernel(const float* __restrict__ partials, int n,
                               float* __restrict__ out, float scale) {
    const int lane = threadIdx.x;
    float s = 0.f;
    for (int i = lane; i < n; i += 32) s += partials[i];
#pragma unroll
    for (int off = 16; off; off >>= 1) s += __shfl_xor(s, off, 32);
    if (lane == 0) out[0] = s * scale;
}

extern "C" void kernel_launch(void* const* d_in, const int* in_sizes, int n_in,
                              void* d_out, int out_size, void* d_ws, size_t ws_size,
                              hipStream_t stream) {
    (void)n_in; (void)out_size; (void)ws_size;
    const float* z  = (const float*)d_in[0];     // (B, 64)
    const float* cb = (const float*)d_in[1];     // (K, 64)
    const int Bn = in_sizes[0] / 64;
    const int Kn = in_sizes[1] / 64;

    float* csq      = (float*)d_ws;              // K floats
    float* partials = csq + Kn;                  // Bn/128 floats

    float* out_zq   = (float*)d_out;             // B*64
    float* out_idx  = out_zq + (size_t)Bn * 64;  // B (indices as float)
    float* out_loss = out_idx + Bn;              // 1

    const int nblk = Bn / 128;

    vq_csq_kernel<<<(Kn + 7) / 8, 256, 0, stream>>>(cb, csq, Kn);
    vq_main_kernel<<<nblk, 256, 0, stream>>>(z, cb, csq, out_zq, out_idx, partials, Kn);
    vq_loss_kernel<<<1, 32, 0, stream>>>(partials, nblk, out_loss,
                                         1.0f / ((float)Bn * 64.0f));
}